// EnhancedFoodDrugGNN_352187318508
// MI455X (gfx1250) — compile-verified
//
#include <hip/hip_runtime.h>
#include <hip/hip_bf16.h>

// ---------------------------------------------------------------------------
// 3-layer GCN forward for MI455X (gfx1250, wave32).
// GEMM via V_WMMA_F32_16X16X4_F32 (exact fp32 matrix path).
// Scatter via wave-per-edge float4 atomic adds (bandwidth-bound part).
// ---------------------------------------------------------------------------

typedef float v2f __attribute__((ext_vector_type(2)));
typedef float v8f __attribute__((ext_vector_type(8)));

#define DIM 128
#define BN_EPS 1e-5f

// ---- degree / norm ---------------------------------------------------------

__global__ void k_deg_init(float* __restrict__ deg, int n) {
    int i = blockIdx.x * blockDim.x + threadIdx.x;
    if (i < n) deg[i] = 1.0f;   // self-loop
}

__global__ void k_deg_edges(const int* __restrict__ dst, float* __restrict__ deg, int e) {
    int i = blockIdx.x * blockDim.x + threadIdx.x;
    if (i < e) atomicAdd(&deg[dst[i]], 1.0f);
}

__global__ void k_deg_finish(float* __restrict__ deg, int n) {
    int i = blockIdx.x * blockDim.x + threadIdx.x;
    if (i < n) deg[i] = __frsqrt_rn(deg[i]);   // deg >= 1 always
}

// ---- GEMM: XW = H @ W, H:[n,128], W:[128,128], via WMMA f32 16x16x4 --------
// One wave computes a 16-row x 128-col strip: 8 accumulators, K stepped by 4.
// A 16x4 layout : lanes 0-15 hold M=lane, {K, K+1}; lanes 16-31 hold {K+2, K+3}.
// B 4x16 layout : lanes 0-15 hold N=lane, rows {K, K+1}; lanes 16-31 rows {K+2,K+3}.
// C/D 16x16     : VGPR j, lanes 0-15 -> M=j, lanes 16-31 -> M=j+8; N = lane%16.

__global__ __launch_bounds__(256) void k_gemm_wmma(const float* __restrict__ H,
                                                   const float* __restrict__ Wm,
                                                   float* __restrict__ XW, int n_rows) {
    int wave = (blockIdx.x * blockDim.x + threadIdx.x) >> 5;
    int lane = threadIdx.x & 31;
    int m0 = wave * 16;
    if (m0 >= n_rows) return;

    int lhalf = lane >> 4;      // 0 or 1
    int lmod  = lane & 15;

    v8f acc[8];
    #pragma unroll
    for (int t = 0; t < 8; ++t)
        #pragma unroll
        for (int j = 0; j < 8; ++j) acc[t][j] = 0.0f;

    const float* hrow = H + (size_t)(m0 + lmod) * DIM;

    for (int k = 0; k < DIM; k += 4) {
        v2f a;
        a.x = hrow[k + 2 * lhalf];
        a.y = hrow[k + 2 * lhalf + 1];
        #pragma unroll
        for (int t = 0; t < 8; ++t) {
            int n0 = t * 16;
            v2f b;
            b.x = Wm[(size_t)(k + 2 * lhalf)     * DIM + n0 + lmod];
            b.y = Wm[(size_t)(k + 2 * lhalf + 1) * DIM + n0 + lmod];
            acc[t] = __builtin_amdgcn_wmma_f32_16x16x4_f32(
                false, a, false, b, (short)0, acc[t], false, false);
        }
    }

    // store: row = m0 + j + 8*lhalf, col = n0 + lmod
    #pragma unroll
    for (int t = 0; t < 8; ++t) {
        float* orow = XW + (size_t)(m0 + 8 * lhalf) * DIM + t * 16 + lmod;
        #pragma unroll
        for (int j = 0; j < 8; ++j)
            orow[(size_t)j * DIM] = acc[t][j];
    }
}

// ---- out = bias + dinv^2 * xw (self loop) [+ residual h] -------------------

__global__ void k_init_out(const float* __restrict__ XW, const float* __restrict__ H,
                           const float* __restrict__ dinv, const float* __restrict__ bias,
                           float* __restrict__ OUT, int n, int add_resid) {
    long long idx = (long long)blockIdx.x * blockDim.x + threadIdx.x;
    long long total = (long long)n * DIM;
    if (idx >= total) return;
    int node = (int)(idx >> 7);
    int d    = (int)(idx & (DIM - 1));
    float di = dinv[node];
    float v = bias[d] + di * di * XW[idx];
    if (add_resid) v += H[idx];
    OUT[idx] = v;
}

// ---- scatter: wave per edge, 4 floats per lane -----------------------------

__global__ __launch_bounds__(256) void k_scatter(const float* __restrict__ XW,
                                                 const int* __restrict__ src,
                                                 const int* __restrict__ dst,
                                                 const float* __restrict__ dinv,
                                                 float* __restrict__ OUT, int n_edges) {
    long long t = (long long)blockIdx.x * blockDim.x + threadIdx.x;
    int e    = (int)(t >> 5);
    int lane = (int)(t & 31);
    if (e >= n_edges) return;
    int s = src[e];
    int d = dst[e];
    float w = dinv[s] * dinv[d];
    const float4* xs = (const float4*)(XW + (size_t)s * DIM);
    float4 v = xs[lane];                       // 32 lanes x 4 = 128 floats
    float* od = OUT + (size_t)d * DIM + lane * 4;
    atomicAdd(od + 0, w * v.x);
    atomicAdd(od + 1, w * v.y);
    atomicAdd(od + 2, w * v.z);
    atomicAdd(od + 3, w * v.w);
}

// ---- BatchNorm -------------------------------------------------------------

__global__ void k_zero_stats(float* __restrict__ stats) {
    int i = threadIdx.x;
    if (i < 2 * DIM) stats[i] = 0.0f;
}

#define BN_ROWS 256

__global__ __launch_bounds__(DIM) void k_bn_stats(const float* __restrict__ OUT,
                                                  float* __restrict__ stats, int n) {
    int d  = threadIdx.x;              // column, coalesced across the block
    int r0 = blockIdx.x * BN_ROWS;
    int r1 = r0 + BN_ROWS; if (r1 > n) r1 = n;
    float s = 0.0f, s2 = 0.0f;
    for (int r = r0; r < r1; ++r) {
        float v = OUT[(size_t)r * DIM + d];
        s += v; s2 += v * v;
    }
    atomicAdd(&stats[d], s);
    atomicAdd(&stats[DIM + d], s2);
}

__global__ __launch_bounds__(DIM) void k_bn_scale(float* __restrict__ stats,
                                                  const float* __restrict__ gamma,
                                                  const float* __restrict__ beta, int n) {
    int d = threadIdx.x;
    float invn = 1.0f / (float)n;
    float mu  = stats[d] * invn;
    float var = stats[DIM + d] * invn - mu * mu;     // biased var
    float sc  = gamma[d] * __frsqrt_rn(var + BN_EPS);
    stats[d]       = sc;                  // scale
    stats[DIM + d] = beta[d] - mu * sc;   // shift
}

__global__ void k_bn_apply(float* __restrict__ OUT, const float* __restrict__ stats, int n) {
    long long idx = (long long)blockIdx.x * blockDim.x + threadIdx.x;
    long long total = (long long)n * DIM;
    if (idx >= total) return;
    int d = (int)(idx & (DIM - 1));
    float v = OUT[idx] * stats[d] + stats[DIM + d];
    OUT[idx] = v > 0.0f ? v : 0.0f;
}

// ---------------------------------------------------------------------------

extern "C" void kernel_launch(void* const* d_in, const int* in_sizes, int n_in,
                              void* d_out, int out_size, void* d_ws, size_t ws_size,
                              hipStream_t stream) {
    const int N = in_sizes[0] / DIM;       // 100000
    const int E = in_sizes[1] / 2;         // 1600000

    const float* x   = (const float*)d_in[0];
    const int*   src = (const int*)d_in[1];
    const int*   dst = src + E;
    const float* Ws[3]     = {(const float*)d_in[2],  (const float*)d_in[3],  (const float*)d_in[4]};
    const float* bs[3]     = {(const float*)d_in[5],  (const float*)d_in[6],  (const float*)d_in[7]};
    const float* gammas[3] = {(const float*)d_in[8],  (const float*)d_in[9],  (const float*)d_in[10]};
    const float* betas[3]  = {(const float*)d_in[11], (const float*)d_in[12], (const float*)d_in[13]};

    float* ws    = (float*)d_ws;
    float* dinv  = ws;                               // N
    float* stats = dinv + N;                         // 256
    float* xw    = stats + 2 * DIM;                  // N*128
    float* bufA  = xw + (size_t)N * DIM;             // N*128
    float* bufB  = bufA + (size_t)N * DIM;           // N*128

    const int B256 = 256;
    long long totalND = (long long)N * DIM;
    int gridN    = (N + B256 - 1) / B256;
    int gridE    = (E + B256 - 1) / B256;
    int gridND   = (int)((totalND + B256 - 1) / B256);
    int gridScat = (int)(((long long)E * 32 + B256 - 1) / B256);
    int waves    = (N + 15) / 16;
    int gridGemm = (waves * 32 + B256 - 1) / B256;
    int gridBN   = (N + BN_ROWS - 1) / BN_ROWS;

    // degree -> dinv
    k_deg_init<<<gridN, B256, 0, stream>>>(dinv, N);
    k_deg_edges<<<gridE, B256, 0, stream>>>(dst, dinv, E);
    k_deg_finish<<<gridN, B256, 0, stream>>>(dinv, N);

    const float* h_ptr  = x;
    float*       out_ptr[3] = {bufA, bufB, (float*)d_out};

    for (int layer = 0; layer < 3; ++layer) {
        float* out = out_ptr[layer];

        k_gemm_wmma<<<gridGemm, B256, 0, stream>>>(h_ptr, Ws[layer], xw, N);
        k_init_out<<<gridND, B256, 0, stream>>>(xw, h_ptr, dinv, bs[layer], out, N,
                                                layer > 0 ? 1 : 0);
        k_scatter<<<gridScat, B256, 0, stream>>>(xw, src, dst, dinv, out, E);

        k_zero_stats<<<1, 2 * DIM, 0, stream>>>(stats);
        k_bn_stats<<<gridBN, DIM, 0, stream>>>(out, stats, N);
        k_bn_scale<<<1, DIM, 0, stream>>>(stats, gammas[layer], betas[layer], N);
        k_bn_apply<<<gridND, B256, 0, stream>>>(out, stats, N);

        h_ptr = out;
    }
}